// MGCN_37271726194960
// MI455X (gfx1250) — compile-verified
//
#include <hip/hip_runtime.h>

#define D    128
#define DV4  32   // D/4 float4 chunks per row

typedef float f4 __attribute__((ext_vector_type(4)));

// ---------------------------------------------------------------------------
// Edge kernel: one wave32 per edge; async-to-LDS double-buffered gathers.
//   HBM -> LDS   : global_load_async_to_lds_b128 (async engine, ASYNCcnt)
//   LDS -> VGPR  : ds_load_b128 (DS pipe, DScnt)
//   VGPR -> HBM  : global_atomic_add_f32 (L2 atomic units)
//   LDS layout per wave: 2 buffers x (x-row[128] | ee-row[128]) floats
//   8 waves/block * 2 * 256 floats = 4096 floats = 16 KB
// ---------------------------------------------------------------------------

__global__ __launch_bounds__(256) void k_edge(
    const int*   __restrict__ eidx,   // [2,E]
    const int*   __restrict__ etype,  // [E]
    const int*   __restrict__ eids,   // [E]
    const float* __restrict__ x,      // [N,D]  (L2-resident, 25.6 MB)
    const float* __restrict__ rel,    // [R,D]  (L2-resident, 256 KB)
    const float* __restrict__ eemb,   // [E,D]  (streamed, 327 MB/conv)
    const float* __restrict__ norm,   // [E]
    float*       __restrict__ agg,    // [N,D]
    int E)
{
  __shared__ __align__(16) float lds[8 * 2 * 2 * (D / 2)];  // 2048 floats
  (void)lds;  // written by async engine, read by ds_load asm

  const int lane  = threadIdx.x & 31;
  const int wib   = threadIdx.x >> 5;                       // wave in block
  const int nwav  = (gridDim.x * blockDim.x) >> 5;          // total waves
  int e = blockIdx.x * (blockDim.x >> 5) + wib;
  if (e >= E) return;

  const int dbase = lane << 2;                              // float index in row

  // issue async gathers of x[src] and edge_emb[id] rows into LDS buffer b;
  // the async engine streams them while this wave issues atomics for the
  // previous edge (ASYNCcnt tracks completion, loads retire in order).
  auto issue = [&](int ei, int b) {
    const int s  = eidx[ei];
    const int id = eids[ei];
    const float* gx = x    + (size_t)s  * D + dbase;
    const float* ge = eemb + (size_t)id * D + dbase;
    const unsigned lx = (unsigned)((wib * 2048) + (b * 1024) + (dbase * 4)); // bytes
    asm volatile("global_load_async_to_lds_b128 %0, %1, off"
                 :: "v"(lx), "v"(gx) : "memory");
    asm volatile("global_load_async_to_lds_b128 %0, %1, off"
                 :: "v"(lx + 512u), "v"(ge) : "memory");
  };

  issue(e, 0);
  int b = 0;
  for (;;) {
    const int  en   = e + nwav;
    const bool more = (en < E);
    if (more) {
      issue(en, b ^ 1);
      // allow the 2 just-issued ops to stay in flight; current buffer done
      asm volatile("s_wait_asynccnt 0x2" ::: "memory");
    } else {
      asm volatile("s_wait_asynccnt 0x0" ::: "memory");
    }

    // consume current buffer through the DS pipe (block-relative LDS byte address)
    f4 xv, ev;
    const unsigned laddr = (unsigned)((wib * 2048) + (b * 1024) + (dbase * 4));
    asm volatile("ds_load_b128 %0, %2\n\t"
                 "ds_load_b128 %1, %2 offset:512\n\t"
                 "s_wait_dscnt 0x0"
                 : "=&v"(xv), "=&v"(ev)
                 : "v"(laddr)
                 : "memory");

    const int    t  = etype[e];
    const int    dn = eidx[E + e];
    const float  nr = norm[e];
    const f4     rv = *(const f4*)(rel + (size_t)t * D + dbase);

    float* ap = agg + (size_t)dn * D + dbase;
    unsafeAtomicAdd(ap + 0, xv.x * ev.x * rv.x * nr);
    unsafeAtomicAdd(ap + 1, xv.y * ev.y * rv.y * nr);
    unsafeAtomicAdd(ap + 2, xv.z * ev.z * rv.z * nr);
    unsafeAtomicAdd(ap + 3, xv.w * ev.w * rv.w * nr);

    if (!more) break;
    e = en; b ^= 1;
  }
}

// ---------------------------------------------------------------------------
// Precompute / elementwise kernels
// ---------------------------------------------------------------------------

__global__ void k_init_degcnt(float* __restrict__ deg, float* __restrict__ cnt, int n) {
  int i = blockIdx.x * blockDim.x + threadIdx.x;
  if (i < n) { deg[i] = 1.0f; cnt[i] = 1.0f; }   // self loop contributes 1 to each
}

__global__ void k_accum_degcnt(const int* __restrict__ eidx,
                               float* __restrict__ deg, float* __restrict__ cnt, int E) {
  int e = blockIdx.x * blockDim.x + threadIdx.x;
  if (e < E) {
    unsafeAtomicAdd(&deg[eidx[e]],     1.0f);   // row 0 = src
    unsafeAtomicAdd(&cnt[eidx[E + e]], 1.0f);   // row 1 = dst
  }
}

__global__ void k_scalars(const float* __restrict__ deg, const float* __restrict__ cnt,
                          float* __restrict__ dinv, float* __restrict__ dinv2,
                          float* __restrict__ invc, int n) {
  int i = blockIdx.x * blockDim.x + threadIdx.x;
  if (i < n) {
    float dv = rsqrtf(deg[i]);       // deg >= 1 always (self loop)
    dinv[i]  = dv;
    dinv2[i] = dv * dv;              // self-loop norm
    invc[i]  = 1.0f / cnt[i];        // cnt >= 1 always
  }
}

__global__ void k_norm(const int* __restrict__ eidx, const float* __restrict__ dinv,
                       float* __restrict__ norm, int E) {
  int e = blockIdx.x * blockDim.x + threadIdx.x;
  if (e < E) norm[e] = dinv[eidx[e]] * dinv[eidx[E + e]];
}

// agg[i,:] = x[i,:] * dinv2[i]   (self-loop message; also initializes agg)
__global__ void k_selfinit(const float* __restrict__ x, const float* __restrict__ dinv2,
                           float* __restrict__ agg, int n) {
  int t = blockIdx.x * blockDim.x + threadIdx.x;
  if (t < n * DV4) {
    int i = t >> 5;                  // t / DV4
    float s = dinv2[i];
    f4 v = ((const f4*)x)[t];
    v.x *= s; v.y *= s; v.z *= s; v.w *= s;
    ((f4*)agg)[t] = v;
  }
}

// y[i,:] = agg[i,:] * invc[i]  (+ optional relu)
__global__ void k_finalize(const float* __restrict__ agg, const float* __restrict__ invc,
                           float* __restrict__ y, int n, int relu) {
  int t = blockIdx.x * blockDim.x + threadIdx.x;
  if (t < n * DV4) {
    int i = t >> 5;
    float s = invc[i];
    f4 v = ((const f4*)agg)[t];
    v.x *= s; v.y *= s; v.z *= s; v.w *= s;
    if (relu) {
      v.x = fmaxf(v.x, 0.0f); v.y = fmaxf(v.y, 0.0f);
      v.z = fmaxf(v.z, 0.0f); v.w = fmaxf(v.w, 0.0f);
    }
    ((f4*)y)[t] = v;
  }
}

// ---------------------------------------------------------------------------
// Launch
// ---------------------------------------------------------------------------

extern "C" void kernel_launch(void* const* d_in, const int* in_sizes, int n_in,
                              void* d_out, int out_size, void* d_ws, size_t ws_size,
                              hipStream_t stream) {
  const int*   eidx  = (const int*)  d_in[0];   // [2,E]
  const int*   etype = (const int*)  d_in[1];   // [E]
  const int*   eids  = (const int*)  d_in[2];   // [E]
  const float* ent   = (const float*)d_in[3];   // [N,D]
  const float* rel   = (const float*)d_in[4];   // [R,D]
  const float* eemb  = (const float*)d_in[5];   // [E,D]
  float*       out   = (float*)d_out;

  const int E = in_sizes[1];
  const int N = in_sizes[3] / D;

  // workspace carve-up (floats): 5N + E + 2*N*D  (~55 MB)
  float* ws    = (float*)d_ws;
  float* deg   = ws;
  float* cnt   = deg   + N;
  float* dinv  = cnt   + N;
  float* dinv2 = dinv  + N;
  float* invc  = dinv2 + N;
  float* norm  = invc  + N;
  float* agg   = norm  + E;
  float* xbuf  = agg   + (size_t)N * D;

  const int T = 256;
  const int nodeBlk  = (N + T - 1) / T;
  const int edgeBlk  = (E + T - 1) / T;
  const int rowBlk   = (N * DV4 + T - 1) / T;
  const int msgBlk   = 2048;                     // 16384 waves, ~39 edges each

  // graph statistics (identical every call; rebuilt deterministically)
  k_init_degcnt <<<nodeBlk, T, 0, stream>>>(deg, cnt, N);
  k_accum_degcnt<<<edgeBlk, T, 0, stream>>>(eidx, deg, cnt, E);
  k_scalars     <<<nodeBlk, T, 0, stream>>>(deg, cnt, dinv, dinv2, invc, N);
  k_norm        <<<edgeBlk, T, 0, stream>>>(eidx, dinv, norm, E);

  // conv1: ent -> xbuf
  k_selfinit<<<rowBlk, T, 0, stream>>>(ent, dinv2, agg, N);
  k_edge    <<<msgBlk, T, 0, stream>>>(eidx, etype, eids, ent, rel, eemb, norm, agg, E);
  k_finalize<<<rowBlk, T, 0, stream>>>(agg, invc, xbuf, N, 0);

  // conv2 (+relu): xbuf -> xbuf
  k_selfinit<<<rowBlk, T, 0, stream>>>(xbuf, dinv2, agg, N);
  k_edge    <<<msgBlk, T, 0, stream>>>(eidx, etype, eids, xbuf, rel, eemb, norm, agg, E);
  k_finalize<<<rowBlk, T, 0, stream>>>(agg, invc, xbuf, N, 1);

  // conv3: xbuf -> out
  k_selfinit<<<rowBlk, T, 0, stream>>>(xbuf, dinv2, agg, N);
  k_edge    <<<msgBlk, T, 0, stream>>>(eidx, etype, eids, xbuf, rel, eemb, norm, agg, E);
  k_finalize<<<rowBlk, T, 0, stream>>>(agg, invc, out, N, 0);
}